// ChebConv_37452114821814
// MI455X (gfx1250) — compile-verified
//
#include <hip/hip_runtime.h>
#include <hip/hip_bf16.h>
#include <stdint.h>

typedef __attribute__((ext_vector_type(2))) float v2f;
typedef __attribute__((ext_vector_type(8))) float v8f;

#define N_NODES 100000
#define N_EDGES 3200000
#define NFEAT   128
#define NKC     6
#define N_TILES (N_NODES / 16)   // 6250 exactly

// ---- monotone float<->uint mapping so unsigned atomicMax orders like float ----
__device__ __forceinline__ unsigned mono_from_f(float f) {
    unsigned u = __float_as_uint(f);
    return (u & 0x80000000u) ? ~u : (u | 0x80000000u);
}
__device__ __forceinline__ float f_from_mono(unsigned u) {
    unsigned b = (u & 0x80000000u) ? (u & 0x7FFFFFFFu) : ~u;
    return __uint_as_float(b);
}

// ---- 0: zero scratch accumulators ----
__global__ void k_zero(float* deg, float* rsum, unsigned* maxbits) {
    int i = blockIdx.x * blockDim.x + threadIdx.x;
    if (i < N_NODES) { deg[i] = 0.0f; rsum[i] = 0.0f; }
    if (i == 0) *maxbits = 0u;   // mono() of -inf side
}

// ---- 1: edge scatter: degree, receiver-sum, and max(-e) via block reduction ----
__global__ void k_edges(const float* __restrict__ edges,
                        const int* __restrict__ snd,
                        const int* __restrict__ rcv,
                        float* deg, float* rsum, unsigned* maxbits) {
    __shared__ unsigned red[256];
    int j = blockIdx.x * blockDim.x + threadIdx.x;   // E divisible by 256
    float e = edges[j];
    atomicAdd(&deg[snd[j]], e);
    atomicAdd(&rsum[rcv[j]], e);
    red[threadIdx.x] = mono_from_f(-e);
    __syncthreads();
    #pragma unroll
    for (int s = 128; s > 0; s >>= 1) {
        if (threadIdx.x < s) {
            unsigned a = red[threadIdx.x], b = red[threadIdx.x + s];
            red[threadIdx.x] = (a > b) ? a : b;
        }
        __syncthreads();
    }
    if (threadIdx.x == 0) atomicMax(maxbits, red[0]);
}

// ---- 2: fold degree into the max (lap_w = concat(-e, deg)) ----
__global__ void k_degmax(const float* __restrict__ deg, unsigned* maxbits) {
    __shared__ unsigned red[256];
    int i = blockIdx.x * blockDim.x + threadIdx.x;
    red[threadIdx.x] = (i < N_NODES) ? mono_from_f(deg[i]) : 0u;
    __syncthreads();
    #pragma unroll
    for (int s = 128; s > 0; s >>= 1) {
        if (threadIdx.x < s) {
            unsigned a = red[threadIdx.x], b = red[threadIdx.x + s];
            red[threadIdx.x] = (a > b) ? a : b;
        }
        __syncthreads();
    }
    if (threadIdx.x == 0) atomicMax(maxbits, red[0]);
}

// ---- 3: s_i = (deg_i - sum_{recv=i} e) / max(lap_w)   (norm = lap_w / lap_w.max()) ----
__global__ void k_scale(const float* __restrict__ deg, const float* __restrict__ rsum,
                        const unsigned* __restrict__ maxbits, float* __restrict__ s) {
    int i = blockIdx.x * blockDim.x + threadIdx.x;
    if (i < N_NODES) {
        float mw = f_from_mono(*maxbits);
        s[i] = (deg[i] - rsum[i]) / mw;
    }
}

// ---- 4: fold the K weight matrices: WAB[k][0:128]=Σ(1-q)Wk_q, WAB[k][128:256]=Σ q·Wk_q ----
__global__ void k_weights(const float* __restrict__ Wk, const float* __restrict__ bk,
                          const float* __restrict__ bias,
                          float* __restrict__ WAB, float* __restrict__ bconst) {
    int idx = blockIdx.x * blockDim.x + threadIdx.x;   // 128 blocks * 256 = 32768
    if (idx < NFEAT * 256) {
        int k = idx >> 8, c = idx & 255;
        float acc = 0.0f;
        if (c < NFEAT) {
            #pragma unroll
            for (int q = 0; q < NKC; ++q)
                acc += (1.0f - (float)q) * Wk[q * NFEAT * NFEAT + k * NFEAT + c];
        } else {
            int cc = c - NFEAT;
            #pragma unroll
            for (int q = 1; q < NKC; ++q)
                acc += (float)q * Wk[q * NFEAT * NFEAT + k * NFEAT + cc];
        }
        WAB[k * 256 + c] = acc;
    }
    if (idx < NFEAT) {
        float b = bias[idx];
        #pragma unroll
        for (int q = 0; q < NKC; ++q) b += bk[q * NFEAT + idx];
        bconst[idx] = b;
    }
}

// ---- 5: fused GEMM: out = nodes@WB + s*(nodes@WA) + bconst, via V_WMMA_F32_16X16X4_F32 ----
__global__ __launch_bounds__(256)
void k_gemm(const float* __restrict__ nodes, const float* __restrict__ WAB,
            const float* __restrict__ bconst, const float* __restrict__ s,
            float* __restrict__ out) {
    __shared__ float ldsA[16 * 132];   // 16x128 A tile, stride 132 -> conflict-free frag reads
    __shared__ float sTile[16];

    const int tid  = threadIdx.x;
    const int wave = tid >> 5;          // 8 waves: wave w owns output cols [16w,16w+16)
    const int lane = tid & 31;
    const int lid  = lane & 15;
    const int half = lane >> 4;
    const int colBase = wave * 16;

    // Register-cache all B fragments for this wave's two chains (WB cols, WA cols).
    // 32 k-steps x 2 chains x v2f = 128 VGPRs; loaded from L2 once per block.
    v2f Bc[32], Ba[32];
    {
        const int cB = colBase + lid;
        const int cA = NFEAT + colBase + lid;
        #pragma unroll
        for (int i = 0; i < 32; ++i) {
            int kr = 4 * i + 2 * half;                 // B frag VGPR v holds K = kr+v
            v2f b0, b1;
            b0.x = WAB[kr * 256 + cB];  b0.y = WAB[(kr + 1) * 256 + cB];
            b1.x = WAB[kr * 256 + cA];  b1.y = WAB[(kr + 1) * 256 + cA];
            Bc[i] = b0; Ba[i] = b1;
        }
    }
    const float bc = bconst[colBase + lid];

    for (int t = blockIdx.x; t < N_TILES; t += gridDim.x) {
        const int rowBase = t * 16;

        // cooperative, coalesced A-tile stage: 512 float4 slots / 256 threads
        #pragma unroll
        for (int p = 0; p < 2; ++p) {
            int idx = tid + p * 256;
            int r   = idx >> 5;
            int c4  = (idx & 31) << 2;
            const float4 v = *(const float4*)(nodes + (size_t)(rowBase + r) * NFEAT + c4);
            *(float4*)(&ldsA[r * 132 + c4]) = v;
        }
        if (tid < 16) sTile[tid] = s[rowBase + tid];
        __syncthreads();

        v8f accB = {};   // YB chain (constant-weight part)
        v8f accA = {};   // YA chain (s-scaled part)
        #pragma unroll
        for (int i = 0; i < 32; ++i) {
            const int k0 = 4 * i + 2 * half;           // A frag VGPR v holds K = k0+v
            v2f a;
            a.x = ldsA[lid * 132 + k0];
            a.y = ldsA[lid * 132 + k0 + 1];
            accB = __builtin_amdgcn_wmma_f32_16x16x4_f32(
                false, a, false, Bc[i], (short)0, accB, false, false);
            accA = __builtin_amdgcn_wmma_f32_16x16x4_f32(
                false, a, false, Ba[i], (short)0, accA, false, false);
        }

        // epilogue: C/D layout — VGPR v is row v+8*half, lane is column
        const int col = colBase + lid;
        #pragma unroll
        for (int v = 0; v < 8; ++v) {
            const int rIn = v + 8 * half;
            out[(size_t)(rowBase + rIn) * NFEAT + col] =
                accB[v] + sTile[rIn] * accA[v] + bc;
        }
        __syncthreads();
    }
}

extern "C" void kernel_launch(void* const* d_in, const int* in_sizes, int n_in,
                              void* d_out, int out_size, void* d_ws, size_t ws_size,
                              hipStream_t stream) {
    const float* nodes     = (const float*)d_in[0];
    const float* edges     = (const float*)d_in[1];
    const int*   senders   = (const int*)d_in[2];
    const int*   receivers = (const int*)d_in[3];
    const float* Wk        = (const float*)d_in[4];
    const float* bk        = (const float*)d_in[5];
    const float* bias      = (const float*)d_in[6];
    float*       out       = (float*)d_out;

    // workspace layout (floats): deg | rsum | s | WAB(128x256) | bconst(128) | maxbits(u32)
    float*    ws      = (float*)d_ws;
    float*    deg     = ws;
    float*    rsum    = ws + N_NODES;
    float*    sbuf    = ws + 2 * N_NODES;
    float*    WAB     = ws + 3 * N_NODES;
    float*    bconst  = WAB + NFEAT * 256;
    unsigned* maxbits = (unsigned*)(bconst + NFEAT);

    const int nb_nodes = (N_NODES + 255) / 256;

    k_zero   <<<nb_nodes,        256, 0, stream>>>(deg, rsum, maxbits);
    k_edges  <<<N_EDGES / 256,   256, 0, stream>>>(edges, senders, receivers, deg, rsum, maxbits);
    k_degmax <<<nb_nodes,        256, 0, stream>>>(deg, maxbits);
    k_scale  <<<nb_nodes,        256, 0, stream>>>(deg, rsum, maxbits, sbuf);
    k_weights<<<NFEAT,           256, 0, stream>>>(Wk, bk, bias, WAB, bconst);
    k_gemm   <<<625,             256, 0, stream>>>(nodes, WAB, bconst, sbuf, out);
}